// DepthwiseCrossViTMAE_1030792151500
// MI455X (gfx1250) — compile-verified
//
#include <hip/hip_runtime.h>
#include <hip/hip_bf16.h>

typedef float v2f __attribute__((ext_vector_type(2)));
typedef float v8f __attribute__((ext_vector_type(8)));
typedef unsigned int v4u __attribute__((ext_vector_type(4)));
typedef unsigned int v8u __attribute__((ext_vector_type(8)));

// Problem constants (reference: B=4, L=1024, C=32, K=64, hidden=2048, 3K=192)
#define CCH   32
#define KD    64
#define K3    192
#define HID   2048   // C*K

// ---------------------------------------------------------------------------
// Tensor Data Mover helpers (CDNA5 ISA ch.8: D# groups 0/1, 2-group form).
// f32 elements (data_size code 2). Issued per-wave; EXEC ignored by TDM.
// ---------------------------------------------------------------------------
__device__ __forceinline__ unsigned lds_addr_of(const void* p) {
    // Generic LDS pointer: addr[31:0] is the workgroup-relative LDS byte addr.
    return (unsigned)(uintptr_t)p;
}

__device__ __forceinline__ void tdm_fill_g0(v4u& g0, unsigned lds, unsigned long long ga) {
    g0[0] = 1u;                                   // count=1, user mode
    g0[1] = lds;                                  // lds_addr (bytes)
    g0[2] = (unsigned)ga;                         // global_addr[31:0]
    g0[3] = (unsigned)(ga >> 32) | (2u << 30);    // global_addr[56:32] | type=2
}

// 1-D contiguous copy of `nelem` f32 (nelem < 65536).
__device__ __forceinline__ void tdm_load_1d(const void* lds_dst, const void* gsrc,
                                            unsigned nelem) {
    v4u g0; v8u g1;
    tdm_fill_g0(g0, lds_addr_of(lds_dst), (unsigned long long)(uintptr_t)gsrc);
    g1[0] = (2u << 16);                           // data_size = 4B
    g1[1] = (nelem & 0xFFFFu) << 16;              // tensor_dim0[15:0]
    g1[2] = (nelem >> 16) | (1u << 16);           // tensor_dim0[31:16] | tensor_dim1=1
    g1[3] = (nelem & 0xFFFFu) << 16;              // tile_dim0
    g1[4] = 1u;                                   // tile_dim1=1, tile_dim2=0
    g1[5] = nelem;                                // tensor_dim0_stride[31:0]
    g1[6] = 0u;                                   // stride0[47:32] | stride1[15:0]
    g1[7] = 0u;                                   // stride1[47:16]
    asm volatile("tensor_load_to_lds %0, %1" :: "s"(g0), "s"(g1) : "memory");
}

// 2-D tile copy: tile1 rows of tile0 f32, row stride `stride0` elements.
// Rows packed contiguously into LDS (no padding).
__device__ __forceinline__ void tdm_load_2d(const void* lds_dst, const void* gsrc,
                                            unsigned tile0, unsigned tile1,
                                            unsigned stride0) {
    v4u g0; v8u g1;
    tdm_fill_g0(g0, lds_addr_of(lds_dst), (unsigned long long)(uintptr_t)gsrc);
    g1[0] = (2u << 16);                           // data_size = 4B
    g1[1] = 0u;                                   // tensor_dim0 = 1<<20 (lo16 = 0)
    g1[2] = 0x10u;                                // tensor_dim0 hi | tensor_dim1 lo16=0
    g1[3] = 0x10u | (tile0 << 16);                // tensor_dim1 hi (1<<20) | tile_dim0
    g1[4] = tile1;                                // tile_dim1 | tile_dim2=0
    g1[5] = stride0;                              // tensor_dim0_stride[31:0]
    g1[6] = 0u;
    g1[7] = 0u;
    asm volatile("tensor_load_to_lds %0, %1" :: "s"(g0), "s"(g1) : "memory");
}

// ---------------------------------------------------------------------------
// Kernel 1: per-channel QKV projection.
//   qkv[tok, c, i] = sum_k x[tok, c*64+k] * Wqkv[c, k, i] + bqkv[c, i]
// One block = (channel, 16-token tile); 12 waves, each one 16x16 output tile
// via 16 chained v_wmma_f32_16x16x4_f32. All LDS staging via TDM.
// ---------------------------------------------------------------------------
__global__ __launch_bounds__(384)
void qkv_wmma_kernel(const float* __restrict__ x,
                     const float* __restrict__ Wqkv,
                     const float* __restrict__ bqkv,
                     float* __restrict__ qkv) {
    __shared__ float xs[16 * KD];        // x tile   [row][k]   4 KB
    __shared__ float wsq[KD * K3];       // Wqkv[c]  [k][i]     48 KB
    __shared__ float bs[K3];

    const int blk  = blockIdx.x;
    const int mt   = blk & 255;          // token tile (4096/16 = 256)
    const int c    = blk >> 8;           // channel
    const int tid  = threadIdx.x;
    const int tok0 = mt * 16;

    if (tid < 32) {                      // wave 0 drives the TDM
        tdm_load_2d(xs, x + (size_t)tok0 * HID + c * KD, KD, 16, HID);
        tdm_load_1d(wsq, Wqkv + (size_t)c * KD * K3, KD * K3);
        tdm_load_1d(bs, bqkv + c * K3, K3);
        __builtin_amdgcn_s_wait_tensorcnt(0);
    }
    __syncthreads();

    const int wave = tid >> 5;           // 0..11 -> output n-tile
    const int lane = tid & 31;
    const int row  = lane & 15;          // M row (A) / N col (B, C)
    const int hl   = lane >> 4;          // lane half
    const int n0   = wave * 16;

    v8f acc = {};
    #pragma unroll
    for (int k = 0; k < KD; k += 4) {
        const int ka = k + hl * 2;
        v2f a, b;
        a.x = xs[row * KD + ka];
        a.y = xs[row * KD + ka + 1];
        b.x = wsq[ka * K3 + n0 + row];
        b.y = wsq[(ka + 1) * K3 + n0 + row];
        acc = __builtin_amdgcn_wmma_f32_16x16x4_f32(
            false, a, false, b, (short)0, acc, false, false);
    }

    const float bias = bs[n0 + row];
    #pragma unroll
    for (int i = 0; i < 8; ++i) {
        const int m = i + 8 * hl;        // C/D layout: VGPR i -> M = i + 8*half
        const size_t off = ((size_t)(tok0 + m) * CCH + c) * K3 + n0 + row;
        qkv[off] = acc[i] + bias;
    }
}

// ---------------------------------------------------------------------------
// Kernel 2: fused criss-cross attention + shared output projection.
// One block per token, 256 threads (8 waves).
//   scores[d,c,e] = q[c,d]*k[e,d]/sqrt(K)  (rank-1 per d)
//   attn = softmax over e ; ctx[d,m] = sum_c attn[d,c,m]*v[c,d]
//   out[m,o] = sum_d ctx^T[m,d]*Wout[d,o] + bout[o]   (WMMA, 8 tiles)
// qkvs layout: [c][192] with q = [0,64), k = [64,128), v = [128,192).
// In-place reuse: q slot <- a = q/sqrt(K); v slot <- w = v/Z.
// ---------------------------------------------------------------------------
__global__ __launch_bounds__(256)
void attn_out_wmma_kernel(const float* __restrict__ qkv,
                          const float* __restrict__ Wout,
                          const float* __restrict__ bout,
                          float* __restrict__ out) {
    __shared__ float qkvs[CCH * K3];     // 24 KB (TDM destination)
    __shared__ float mm[CCH * KD];       // m1[c][d]  8 KB
    __shared__ float ctx[CCH * KD];      // ctx^T[m][d]  8 KB
    __shared__ float wos[KD * KD];       // Wout[d][o]  16 KB (TDM destination)
    __shared__ float kmax[KD], kmin[KD];

    const int tok = blockIdx.x;
    const int tid = threadIdx.x;
    const float inv_s = 0.125f;          // 1/sqrt(64)

    if (tid < 32) {
        tdm_load_1d(qkvs, qkv + (size_t)tok * CCH * K3, CCH * K3);
        tdm_load_1d(wos, Wout, KD * KD);
        __builtin_amdgcn_s_wait_tensorcnt(0);
    }
    __syncthreads();

    // per-d key extrema (softmax max via rank-1 structure)
    if (tid < KD) {
        float mx = qkvs[KD + tid], mn = mx;
        #pragma unroll
        for (int e = 1; e < CCH; ++e) {
            const float t = qkvs[e * K3 + KD + tid];
            mx = fmaxf(mx, t);
            mn = fminf(mn, t);
        }
        kmax[tid] = mx;
        kmin[tid] = mn;
    }
    __syncthreads();

    // phase 2: per (c,d) compute a = q/s, m1, w = v/Z  (in-place reuse)
    for (int p = tid; p < CCH * KD; p += 256) {
        const int cc = p >> 6, d = p & 63;
        const int qi = cc * K3 + d;      // q slot
        const float a  = qkvs[qi] * inv_s;
        const float m1 = (a >= 0.f) ? a * kmax[d] : a * kmin[d];
        float z = 0.f;
        #pragma unroll
        for (int e = 0; e < CCH; ++e)
            z += __expf(a * qkvs[e * K3 + KD + d] - m1);
        qkvs[qi] = a;                    // a overwrites q
        mm[p]    = m1;
        qkvs[qi + 2 * KD] = qkvs[qi + 2 * KD] / z;   // w overwrites v
    }
    __syncthreads();

    // phase 3: ctx^T[m][d] = sum_c w[c,d] * exp(a[c,d]*k[m,d] - m1[c,d])
    for (int p = tid; p < KD * CCH; p += 256) {
        const int d = p >> 5, m = p & 31;
        const float kv = qkvs[m * K3 + KD + d];
        float acc = 0.f;
        #pragma unroll
        for (int cc = 0; cc < CCH; ++cc) {
            const int qi = cc * K3 + d;
            acc += qkvs[qi + 2 * KD] * __expf(qkvs[qi] * kv - mm[cc * KD + d]);
        }
        ctx[m * KD + d] = acc;
    }
    __syncthreads();

    // phase 4: out-projection, (32 x 64) @ (64 x 64), 8 waves x one 16x16 tile
    const int wave  = tid >> 5;
    const int lane  = tid & 31;
    const int row   = lane & 15;
    const int hl    = lane >> 4;
    const int rtile = wave & 1;          // M block (channels m)
    const int ntile = wave >> 1;         // N block (output dims o), 0..3

    v8f acc = {};
    #pragma unroll
    for (int k = 0; k < KD; k += 4) {
        const int ka = k + hl * 2;
        v2f a, b;
        a.x = ctx[(rtile * 16 + row) * KD + ka];
        a.y = ctx[(rtile * 16 + row) * KD + ka + 1];
        b.x = wos[ka * KD + ntile * 16 + row];
        b.y = wos[(ka + 1) * KD + ntile * 16 + row];
        acc = __builtin_amdgcn_wmma_f32_16x16x4_f32(
            false, a, false, b, (short)0, acc, false, false);
    }

    const float bias = bout[ntile * 16 + row];
    float* obase = out + (size_t)tok * HID;
    #pragma unroll
    for (int i = 0; i < 8; ++i) {
        const int m = rtile * 16 + i + 8 * hl;
        obase[m * KD + ntile * 16 + row] = acc[i] + bias;
    }
}

// ---------------------------------------------------------------------------
extern "C" void kernel_launch(void* const* d_in, const int* in_sizes, int n_in,
                              void* d_out, int out_size, void* d_ws, size_t ws_size,
                              hipStream_t stream) {
    const float* x    = (const float*)d_in[0];   // (B, L, 2048)
    const float* Wqkv = (const float*)d_in[1];   // (32, 64, 192)
    const float* bqkv = (const float*)d_in[2];   // (32, 192)
    const float* Wout = (const float*)d_in[3];   // (64, 64)
    const float* bout = (const float*)d_in[4];   // (64,)
    float* out = (float*)d_out;

    const int tokens = in_sizes[0] / HID;        // B*L = 4096
    float* qkv_ws = (float*)d_ws;                // tokens*32*192 f32 = ~100 MB

    // Kernel 1: (tokens/16) token tiles * 32 channels, 384 threads/block
    const int mtiles = tokens / 16;
    dim3 g1(mtiles * CCH);
    qkv_wmma_kernel<<<g1, 384, 0, stream>>>(x, Wqkv, bqkv, qkv_ws);

    // Kernel 2: one block per token
    dim3 g2(tokens);
    attn_out_wmma_kernel<<<g2, 256, 0, stream>>>(qkv_ws, Wout, bout, out);
}